// GraphSAGE_7164005450409
// MI455X (gfx1250) — compile-verified
//
#include <hip/hip_runtime.h>
#include <cstdint>
#include <cstddef>

typedef __attribute__((ext_vector_type(2))) float v2f;
typedef __attribute__((ext_vector_type(8))) float v8f;

#define FEAT_D 64

// ---------------------------------------------------------------------------
// Zero-fill (graph-capture safe replacement for memset)
// ---------------------------------------------------------------------------
__global__ void fill_zero(float* __restrict__ p, long n) {
    long i = (long)blockIdx.x * blockDim.x + threadIdx.x;
    const long stride = (long)gridDim.x * blockDim.x;
    for (; i < n; i += stride) p[i] = 0.0f;
}

// ---------------------------------------------------------------------------
// Edge scatter: agg[dst] += feat[src], optionally cnt[dst] += 1.
// 16 threads per edge (2 edges per wave32). Only the group leader (lane 0 /
// lane 16) loads the int64 src/dst pair; it is broadcast to the other 15
// lanes with a wave shuffle — cuts index traffic 16x vs. naive.
// Feature table (12.8 MB) is L2-resident on MI455X (192 MB L2), so the
// gather and the f32 atomics run at L2 speed.
// ---------------------------------------------------------------------------
__global__ void scatter_feat(const float* __restrict__ feat,
                             const long long* __restrict__ src,
                             const long long* __restrict__ dst,
                             float* __restrict__ agg,
                             float* __restrict__ cnt,   // may be nullptr
                             int nE) {
    const int t    = blockIdx.x * blockDim.x + threadIdx.x;
    const int lane = threadIdx.x & 31;
    const int e    = t >> 4;
    const bool leader = (lane & 15) == 0;

    int s = 0, d = 0;
    if (e < nE && leader) {
        s = (int)src[e];
        d = (int)dst[e];
        if (cnt) atomicAdd(&cnt[d], 1.0f);      // fused degree count (layer 0)
    }
    s = __shfl(s, lane & 16, 32);               // broadcast within 16-lane group
    d = __shfl(d, lane & 16, 32);
    if (e >= nE) return;

    const int c = (lane & 15) << 2;             // feature offset, 4 floats
    const float4 v = *reinterpret_cast<const float4*>(feat + (size_t)s * FEAT_D + c);
    float* o = agg + (size_t)d * FEAT_D + c;
    atomicAdd(o + 0, v.x);
    atomicAdd(o + 1, v.y);
    atomicAdd(o + 2, v.z);
    atomicAdd(o + 3, v.w);
}

// ---------------------------------------------------------------------------
// out[16 rows x 64 cols] = (agg/deg) @ Wl^T + x @ Wr^T + bl   (+ optional ReLU)
// One wave per 16-row tile, full-precision V_WMMA_F32_16X16X4_F32 chained
// over K=64 (16 steps), 4 N-tiles held in 4 x v8f accumulators.
//
// Both weight matrices are staged once per block into LDS in *B-fragment
// order*: entry idx = (s*4 + j)*32 + lane holds the v2f that `lane` needs for
// k-step s, n-tile j. Inner-loop ds_load_b64 is then fully coalesced across
// lanes (consecutive 8B -> all 64 banks, conflict-free).
//
// A-fragment layout (32-bit A 16x4, ISA 7.12.2): lanes 0-15 hold row M=lane,
// K = {k0, k0+1}; lanes 16-31 hold row M=lane-16, K = {k0+2, k0+3}.
// C/D: VGPR r -> M = r (lanes 0-15) or r+8 (lanes 16-31), N = lane%16.
// ---------------------------------------------------------------------------
__global__ __launch_bounds__(256)
void sage_linear_wmma(const float* __restrict__ agg,
                      const float* __restrict__ cnt,
                      const float* __restrict__ xin,
                      const float* __restrict__ Wl,
                      const float* __restrict__ bl,
                      const float* __restrict__ Wr,
                      float* __restrict__ out,
                      int nNodes, int applyRelu) {
    __shared__ v2f ldsWl[2048];                  // 16 KB
    __shared__ v2f ldsWr[2048];                  // 16 KB

    const int lane = threadIdx.x & 31;
    const int wave = threadIdx.x >> 5;

    // ---- stage weights into LDS in B-fragment order (all waves help) ----
#pragma unroll
    for (int i = 0; i < 8; ++i) {
        const int idx = threadIdx.x + 256 * i;   // 0..2047
        const int l   = idx & 31;
        const int j   = (idx >> 5) & 3;          // n-tile
        const int s   = idx >> 7;                // k-step
        const int nL  = l & 15;
        const int kh  = (l >> 4) << 1;
        const int off = (j * 16 + nL) * FEAT_D + s * 4 + kh;
        ldsWl[idx] = *reinterpret_cast<const v2f*>(Wl + off);
        ldsWr[idx] = *reinterpret_cast<const v2f*>(Wr + off);
    }
    __syncthreads();

    const int tile = blockIdx.x * 8 + wave;
    const int rowBase = tile << 4;
    if (rowBase >= nNodes) return;               // wave-uniform: EXEC all-1s

    const int mrow  = lane & 15;                 // A row within tile
    const int kHalf = (lane >> 4) << 1;          // 0 or 2 (K sub-column pair)
    const int nLane = lane & 15;                 // N within n-tile

    const float invc = 1.0f / fmaxf(cnt[rowBase + mrow], 1.0f);
    const float* aggRow = agg + (size_t)(rowBase + mrow) * FEAT_D;
    const float* xRow   = xin + (size_t)(rowBase + mrow) * FEAT_D;

    v8f acc[4];
#pragma unroll
    for (int j = 0; j < 4; ++j) acc[j] = (v8f){0, 0, 0, 0, 0, 0, 0, 0};

    for (int s = 0; s < 16; ++s) {
        const int ka = (s << 2) + kHalf;
        v2f aA = *reinterpret_cast<const v2f*>(aggRow + ka);
        aA *= invc;                              // fold mean into A fragment
        const v2f aX = *reinterpret_cast<const v2f*>(xRow + ka);
#pragma unroll
        for (int j = 0; j < 4; ++j) {
            const v2f bL = ldsWl[(s * 4 + j) * 32 + lane];
            const v2f bR = ldsWr[(s * 4 + j) * 32 + lane];
            acc[j] = __builtin_amdgcn_wmma_f32_16x16x4_f32(
                false, aA, false, bL, (short)0, acc[j], false, false);
            acc[j] = __builtin_amdgcn_wmma_f32_16x16x4_f32(
                false, aX, false, bR, (short)0, acc[j], false, false);
        }
    }

    const int rOff = (lane >> 4) << 3;           // 0 or 8 (C/D row half)
#pragma unroll
    for (int j = 0; j < 4; ++j) {
        const float b = bl[j * 16 + nLane];
#pragma unroll
        for (int r = 0; r < 8; ++r) {
            float v = acc[j][r] + b;
            if (applyRelu) v = fmaxf(v, 0.0f);
            out[(size_t)(rowBase + r + rOff) * FEAT_D + (j * 16 + nLane)] = v;
        }
    }
}

// ---------------------------------------------------------------------------
// Launch: zero -> scatter(x, +count) -> gemm0(ReLU) -> zero -> scatter(h)
//         -> gemm1 -> d_out
// ---------------------------------------------------------------------------
extern "C" void kernel_launch(void* const* d_in, const int* in_sizes, int n_in,
                              void* d_out, int out_size, void* d_ws, size_t ws_size,
                              hipStream_t stream) {
    const float*     x   = (const float*)d_in[0];
    const long long* ei  = (const long long*)d_in[1];   // int64 edge_index [2,E]
    const float*     Wl0 = (const float*)d_in[2];
    const float*     bl0 = (const float*)d_in[3];
    const float*     Wr0 = (const float*)d_in[4];
    const float*     Wl1 = (const float*)d_in[5];
    const float*     bl1 = (const float*)d_in[6];
    const float*     Wr1 = (const float*)d_in[7];

    const int nNodes = in_sizes[0] / FEAT_D;            // 50000
    const int nE     = in_sizes[1] / 2;                 // 800000
    const long long* src = ei;
    const long long* dst = ei + nE;

    // Workspace layout (floats): agg[N*64] | cnt[N] | h[N*64]  (~25.8 MB)
    float* agg = (float*)d_ws;
    float* cnt = agg + (size_t)nNodes * FEAT_D;
    float* h   = cnt + nNodes;

    const int scatterThreads = nE * 16;
    const int sblocks = (scatterThreads + 255) / 256;
    const int tiles   = (nNodes + 15) / 16;
    const int gblocks = (tiles + 7) / 8;

    // Layer 0 (scatter also accumulates degree counts into cnt)
    fill_zero<<<1024, 256, 0, stream>>>(agg, (long)nNodes * FEAT_D + nNodes);
    scatter_feat<<<sblocks, 256, 0, stream>>>(x, src, dst, agg, cnt, nE);
    sage_linear_wmma<<<gblocks, 256, 0, stream>>>(agg, cnt, x, Wl0, bl0, Wr0,
                                                  h, nNodes, 1);
    // Layer 1 (reuse cnt)
    fill_zero<<<1024, 256, 0, stream>>>(agg, (long)nNodes * FEAT_D);
    scatter_feat<<<sblocks, 256, 0, stream>>>(h, src, dst, agg, nullptr, nE);
    sage_linear_wmma<<<gblocks, 256, 0, stream>>>(agg, cnt, h, Wl1, bl1, Wr1,
                                                  (float*)d_out, nNodes, 0);
}